// MaxPool4d_31190052504126
// MI455X (gfx1250) — compile-verified
//
#include <hip/hip_runtime.h>

// Fused 4D max pool, K=2 S=2 P=0 over [B=2,C=16,T=32,D=32,H=32,W=32] fp32.
// Non-overlapping windows -> each input byte read exactly once; pure
// HBM-bandwidth kernel (~136 MiB total traffic, ~5.8 us roofline @ 23.3 TB/s).
//
// Per thread: 8x nontemporal global_load_b128 (one per kt,kd,kh plane) ->
// tree max -> pairwise max along W -> 1x nontemporal global_store_b64
// covering 2 adjacent output W elements.
//
// Addressing: all offsets kept in 32 bits (input is 2^27 bytes) so the
// backend emits saddr-form global loads (SGPR base + u32 voffset + imm24),
// and the output flat float2-index is exactly `tid`, so the store stream is
// byte-contiguous across the whole dispatch.

typedef __attribute__((ext_vector_type(4))) float v4f;
typedef __attribute__((ext_vector_type(2))) float v2f;

__global__ __launch_bounds__(256) void maxpool4d_k2s2_kernel(
    const float* __restrict__ in, float* __restrict__ out, unsigned n_threads)
{
    const unsigned tid = blockIdx.x * 256u + threadIdx.x;
    if (tid >= n_threads) return;

    // tid bit layout: [19:15]=bc [14:11]=t [10:7]=d [6:3]=h [2:0]=w0
    // Input base index in float4 units:
    //   idx = w0 | h<<4 | d<<9 | t<<14 | bc<<18
    // i.e. tid with a zero bit inserted at bit positions 3, 8, 13.
    const unsigned idx = (tid & 0x7u)
                       | ((tid & 0x78u)   << 1)
                       | ((tid & 0x780u)  << 2)
                       | ((tid & 0x7800u) << 3)
                       | ((tid >> 15)     << 18);

    const v4f* __restrict__ p = (const v4f*)in;

    // float4-unit strides: kh -> 8, kd -> 256, kt -> 8192 (imm24-foldable)
    v4f a0 = __builtin_nontemporal_load(p + idx);           // kt0 kd0 kh0
    v4f a1 = __builtin_nontemporal_load(p + idx + 8);       // kt0 kd0 kh1
    v4f a2 = __builtin_nontemporal_load(p + idx + 256);     // kt0 kd1 kh0
    v4f a3 = __builtin_nontemporal_load(p + idx + 264);     // kt0 kd1 kh1
    v4f a4 = __builtin_nontemporal_load(p + idx + 8192);    // kt1 kd0 kh0
    v4f a5 = __builtin_nontemporal_load(p + idx + 8200);    // kt1 kd0 kh1
    v4f a6 = __builtin_nontemporal_load(p + idx + 8448);    // kt1 kd1 kh0
    v4f a7 = __builtin_nontemporal_load(p + idx + 8456);    // kt1 kd1 kh1

    // Tree reduction across the 8 (kt,kd,kh) planes
    v4f m01 = __builtin_elementwise_max(a0, a1);
    v4f m23 = __builtin_elementwise_max(a2, a3);
    v4f m45 = __builtin_elementwise_max(a4, a5);
    v4f m67 = __builtin_elementwise_max(a6, a7);
    v4f mA  = __builtin_elementwise_max(m01, m23);
    v4f mB  = __builtin_elementwise_max(m45, m67);
    v4f m   = __builtin_elementwise_max(mA, mB);

    // Pairwise max along W (kw dimension)
    v2f r;
    r.x = fmaxf(m.x, m.y);
    r.y = fmaxf(m.z, m.w);

    // Output flat index in float2 units == tid (perfectly contiguous stores)
    __builtin_nontemporal_store(r, (v2f*)out + tid);
}

extern "C" void kernel_launch(void* const* d_in, const int* in_sizes, int n_in,
                              void* d_out, int out_size, void* d_ws, size_t ws_size,
                              hipStream_t stream)
{
    (void)in_sizes; (void)n_in; (void)d_ws; (void)ws_size;
    const float* x = (const float*)d_in[0];
    float* out = (float*)d_out;

    // 2 outputs per thread
    const unsigned n_threads = (unsigned)(out_size / 2); // 1,048,576 here
    const unsigned blocks = (n_threads + 255u) / 256u;   // 4096 blocks, 8 wave32s each
    maxpool4d_k2s2_kernel<<<blocks, 256, 0, stream>>>(x, out, n_threads);
}